// GCN_49074296324573
// MI455X (gfx1250) — compile-verified
//
#include <hip/hip_runtime.h>
#include <hip/hip_bf16.h>

typedef __attribute__((ext_vector_type(2))) float v2f;
typedef __attribute__((ext_vector_type(8))) float v8f;

#define N_NODES 100000
#define N_EDGES 1600000
#define D_FEAT 64

// ---------------------------------------------------------------------------
// Kernel 1: zero accumulators (accum[N*64] followed contiguously by deg[N]).
// ---------------------------------------------------------------------------
__global__ __launch_bounds__(256) void gcn_zero_kernel(float* __restrict__ p, int n) {
    int i = blockIdx.x * blockDim.x + threadIdx.x;
    if (i < n) p[i] = 0.0f;
}

// ---------------------------------------------------------------------------
// Kernel 2: t = feature @ W^T using V_WMMA_F32_16X16X4_F32 (exact fp32).
// Block = 256 threads (8 waves). Block covers 128 nodes x 64 outputs.
// Wave w computes nodes [base + 16w, base + 16w + 16) x all 64 outputs
// as 4 C-tiles of 16x16, K-loop of 16 steps of K=4.
//
// Fragment layouts (ISA 7.12.2, 32-bit):
//   A 16x4 : lane l, vgpr v ->  M = l&15,  K = (l>>4)*2 + v
//   B 4x16 : lane l, vgpr v ->  N = l&15,  K = (l>>4)*2 + v   (B row = K)
//   C 16x16: lane l, vgpr r ->  M = r + (l>>4)*8, N = l&15
// ---------------------------------------------------------------------------
__global__ __launch_bounds__(256) void gcn_transform_kernel(
    const float* __restrict__ feat, const float* __restrict__ W,
    float* __restrict__ t, int n_nodes)
{
    __shared__ float As[128 * 64];   // 32 KB: 128-node feature tile
    __shared__ float Ws[64 * 64];    // 16 KB: full weight matrix

    const int tid  = threadIdx.x;
    const int wave = tid >> 5;
    const int lane = tid & 31;
    const int ln   = lane & 15;      // column / row-within-half
    const int kh   = lane >> 4;      // K half select
    const int base = blockIdx.x * 128;

    // Stage W: 4096 floats as float4, coalesced.
    {
        const float4* Wv = (const float4*)W;
        float4* Wsv = (float4*)Ws;
        #pragma unroll
        for (int i = tid; i < (64 * 64) / 4; i += 256) Wsv[i] = Wv[i];
    }
    // Stage A tile: 128 rows x 16 float4, coalesced; clamp OOB rows (stores guarded).
    {
        const float4* Fv = (const float4*)feat;
        float4* Asv = (float4*)As;
        #pragma unroll
        for (int i = tid; i < (128 * 64) / 4; i += 256) {
            int node = base + (i >> 4);
            int c4   = i & 15;
            int cn   = node < n_nodes ? node : (n_nodes - 1);
            Asv[i] = Fv[(size_t)cn * 16 + c4];
        }
    }
    __syncthreads();

    v8f c0 = {}, c1 = {}, c2 = {}, c3 = {};
    const int arow = wave * 16 + ln;

    #pragma unroll
    for (int k = 0; k < 16; ++k) {
        const int kc = k * 4 + kh * 2;                       // even -> 8B aligned
        v2f a  = *(const v2f*)&As[arow * 64 + kc];
        v2f b0 = *(const v2f*)&Ws[(0 * 16 + ln) * 64 + kc];  // B[N][K] = W[out][in]
        v2f b1 = *(const v2f*)&Ws[(1 * 16 + ln) * 64 + kc];
        v2f b2 = *(const v2f*)&Ws[(2 * 16 + ln) * 64 + kc];
        v2f b3 = *(const v2f*)&Ws[(3 * 16 + ln) * 64 + kc];
        c0 = __builtin_amdgcn_wmma_f32_16x16x4_f32(false, a, false, b0, (short)0, c0, false, false);
        c1 = __builtin_amdgcn_wmma_f32_16x16x4_f32(false, a, false, b1, (short)0, c1, false, false);
        c2 = __builtin_amdgcn_wmma_f32_16x16x4_f32(false, a, false, b2, (short)0, c2, false, false);
        c3 = __builtin_amdgcn_wmma_f32_16x16x4_f32(false, a, false, b3, (short)0, c3, false, false);
    }

    #pragma unroll
    for (int r = 0; r < 8; ++r) {
        int node = base + wave * 16 + r + kh * 8;
        if (node < n_nodes) {
            float* o = t + (size_t)node * 64;
            o[ 0 + ln] = c0[r];
            o[16 + ln] = c1[r];
            o[32 + ln] = c2[r];
            o[48 + ln] = c3[r];
        }
    }
}

// ---------------------------------------------------------------------------
// Kernel 3: edge scatter. One wave per edge: 32 lanes read the 256B
// transformed source row (float2/lane) and fp32-atomic-add into accum[dst].
// Lane 0 bumps deg[dst]. All traffic is L2-resident (25.6MB tables).
// ---------------------------------------------------------------------------
__global__ __launch_bounds__(256) void gcn_scatter_kernel(
    const float* __restrict__ t,
    const int* __restrict__ esrc, const int* __restrict__ edst,
    float* __restrict__ accum, float* __restrict__ deg, int n_edges)
{
    int wv   = (blockIdx.x * blockDim.x + threadIdx.x) >> 5;
    int lane = threadIdx.x & 31;
    if (wv >= n_edges) return;

    int s = __builtin_amdgcn_readfirstlane(esrc[wv]);   // uniform -> SGPR
    int d = __builtin_amdgcn_readfirstlane(edst[wv]);

    const float2* row = (const float2*)(t + (size_t)s * 64);
    float2 v = row[lane];                               // full 256B row per wave

    float* dst = accum + (size_t)d * 64 + lane * 2;
#if defined(__has_builtin) && __has_builtin(__builtin_amdgcn_global_atomic_fadd_f32)
    __builtin_amdgcn_global_atomic_fadd_f32(dst + 0, v.x);
    __builtin_amdgcn_global_atomic_fadd_f32(dst + 1, v.y);
    if (lane == 0) __builtin_amdgcn_global_atomic_fadd_f32(&deg[d], 1.0f);
#else
    unsafeAtomicAdd(dst + 0, v.x);
    unsafeAtomicAdd(dst + 1, v.y);
    if (lane == 0) unsafeAtomicAdd(&deg[d], 1.0f);
#endif
}

// ---------------------------------------------------------------------------
// Kernel 4: out = relu(accum/deg + b), deg==0 -> relu(b). float4 coalesced.
// ---------------------------------------------------------------------------
__global__ __launch_bounds__(256) void gcn_finalize_kernel(
    const float* __restrict__ accum, const float* __restrict__ deg,
    const float* __restrict__ bias, float* __restrict__ out, int n_nodes)
{
    int i = blockIdx.x * blockDim.x + threadIdx.x;      // one float4 per thread
    int total = n_nodes * 16;
    if (i >= total) return;
    int node = i >> 4;
    float dg = deg[node];
    float inv = dg > 0.0f ? 1.0f / dg : 0.0f;
    float4 a = ((const float4*)accum)[i];
    float4 b = ((const float4*)bias)[i & 15];
    float4 r;
    r.x = fmaxf(a.x * inv + b.x, 0.0f);
    r.y = fmaxf(a.y * inv + b.y, 0.0f);
    r.z = fmaxf(a.z * inv + b.z, 0.0f);
    r.w = fmaxf(a.w * inv + b.w, 0.0f);
    ((float4*)out)[i] = r;
}

extern "C" void kernel_launch(void* const* d_in, const int* in_sizes, int n_in,
                              void* d_out, int out_size, void* d_ws, size_t ws_size,
                              hipStream_t stream) {
    const float* feature = (const float*)d_in[0];   // [100000, 64]
    const int*   esrc    = (const int*)d_in[1];     // [1600000]
    const int*   edst    = (const int*)d_in[2];     // [1600000]
    const float* W       = (const float*)d_in[3];   // [64, 64]
    const float* bias    = (const float*)d_in[4];   // [64]
    float*       out     = (float*)d_out;           // [100000, 64]

    // Workspace layout: accum[N*64] | deg[N] | t[N*64]  (~51.6 MB)
    float* accum = (float*)d_ws;
    float* deg   = accum + (size_t)N_NODES * D_FEAT;
    float* t     = deg + N_NODES;

    // 1) zero accum + deg (contiguous: N*65 floats)
    {
        int n = N_NODES * (D_FEAT + 1);
        gcn_zero_kernel<<<(n + 255) / 256, 256, 0, stream>>>(accum, n);
    }
    // 2) t = feature @ W^T  (WMMA fp32)
    {
        int blocks = (N_NODES + 127) / 128;
        gcn_transform_kernel<<<blocks, 256, 0, stream>>>(feature, W, t, N_NODES);
    }
    // 3) scatter-add transformed messages by destination (one wave per edge)
    {
        int blocks = (N_EDGES + 7) / 8;                 // 8 waves per 256-thread block
        gcn_scatter_kernel<<<blocks, 256, 0, stream>>>(t, esrc, edst, accum, deg, N_EDGES);
    }
    // 4) mean + bias + relu
    {
        int n = N_NODES * 16;
        gcn_finalize_kernel<<<(n + 255) / 256, 256, 0, stream>>>(accum, deg, bias, out, N_NODES);
    }
}